// GraphAutoencoder_17875653886125
// MI455X (gfx1250) — compile-verified
//
#include <hip/hip_runtime.h>
#include <hip/hip_bf16.h>
#include <limits.h>

#define NN  331776        // nodes
#define EE  2654208       // edges (without self loops)
#define EPE (EE + NN)     // edges + self loops
#define BB  4096          // graphs
#define NPG 81            // nodes per graph

typedef __attribute__((ext_vector_type(2))) float v2f;
typedef __attribute__((ext_vector_type(8))) float v8f;

static __host__ int round4(int x)  { return (x + 3) & ~3; }
static __host__ int round16(int x) { return (x + 15) & ~15; }

// ============================================================================
// WMMA fp32 GEMM: H[N, HS] = X[N, CINP] @ Wp[CINP, HS]
//   CINP = round4(Cin)  (X is stored zero-padded at this stride)
//   HS   = round16(Cout) (Wp is stored zero-padded at this stride)
// One wave computes a 16-wide column strip for MT=4 row-tiles of 16.
// All loads/stores unconditional; padded columns of H compute to exactly 0.
// A-frag (ISA 7.12.2, 32-bit A 16x4): lanes 0-15 K=0,1; lanes 16-31 K=2,3.
// ============================================================================
template<int CINP, int HS>
__global__ __launch_bounds__(32) void gemm_wmma(const float* __restrict__ X,
                                                const float* __restrict__ Wp,
                                                float* __restrict__ H)
{
  constexpr int KSTEPS = CINP / 4;
  constexpr int MT = 4;
  const int lane = threadIdx.x;
  const int hw   = lane >> 4;      // half-wave
  const int l    = lane & 15;
  const int m0   = blockIdx.x * (16 * MT);
  const int col  = blockIdx.y * 16 + l;

  // B-frags: loaded once, reused across all MT row tiles
  v2f bf[KSTEPS];
#pragma unroll
  for (int kk = 0; kk < KSTEPS; ++kk) {
    const int ka = kk * 4 + 2 * hw;
    bf[kk].x = Wp[ka * HS + col];
    bf[kk].y = Wp[(ka + 1) * HS + col];
  }

#pragma unroll
  for (int mt = 0; mt < MT; ++mt) {
    const int row = m0 + mt * 16 + l;
    const float* xr = X + (long long)row * CINP + 2 * hw;  // 8B aligned
    v8f acc = {};
#pragma unroll
    for (int kk = 0; kk < KSTEPS; ++kk) {
      v2f a = *(const v2f*)(xr + kk * 4);                  // global_load_b64
      acc = __builtin_amdgcn_wmma_f32_16x16x4_f32(false, a, false, bf[kk],
                                                  (short)0, acc, false, false);
    }
    float* hb = H + (long long)(m0 + mt * 16 + hw * 8) * HS + col;
#pragma unroll
    for (int r = 0; r < 8; ++r) hb[r * HS] = acc[r];
  }
}

// Zero-padded weight pack: Wp[KP][HS] <- W[Cin][Cout]
__global__ void pack_weights(const float* __restrict__ W, float* __restrict__ Wp,
                             int Cin, int Cout, int KP, int HS)
{
  int i = blockIdx.x * blockDim.x + threadIdx.x;
  if (i >= KP * HS) return;
  int k = i / HS, c = i % HS;
  Wp[i] = (k < Cin && c < Cout) ? W[k * Cout + c] : 0.0f;
}

// x0[N,3] -> bufA[N,4] zero padded
__global__ void pack_x(const float* __restrict__ x0, float* __restrict__ xp)
{
  int i = blockIdx.x * blockDim.x + threadIdx.x;
  if (i >= NN * 4) return;
  int c = i & 3;
  xp[i] = (c < 3) ? x0[(i >> 2) * 3 + c] : 0.0f;
}

// es[n] = h[n].a_src ; ed[n] = h[n].a_dst   (H at stride HS)
__global__ void node_scores(const float* __restrict__ H,
                            const float* __restrict__ as,
                            const float* __restrict__ ad,
                            float* __restrict__ es, float* __restrict__ ed,
                            int C, int HS)
{
  int i = blockIdx.x * blockDim.x + threadIdx.x;
  if (i >= NN) return;
  const float* h = H + (long long)i * HS;
  float s = 0.f, d = 0.f;
  for (int c = 0; c < C; ++c) { float v = h[c]; s += v * as[c]; d += v * ad[c]; }
  es[i] = s; ed[i] = d;
}

// out[n][c] = bias[c] (0 in pad cols); m[n] = INT_MIN; denom[n] = 0
__global__ void init_layer(float* __restrict__ out, const float* __restrict__ b,
                           int* __restrict__ m, float* __restrict__ denom,
                           int C, int OS)
{
  long long i = (long long)blockIdx.x * blockDim.x + threadIdx.x;
  long long nc = (long long)NN * OS;
  if (i < nc) {
    int c = (int)(i % OS);
    out[i] = (c < C) ? b[c] : 0.0f;
  }
  if (i < NN) { m[i] = INT_MIN; denom[i] = 0.0f; }
}

__device__ __forceinline__ int f32_order(float f) {
  int i = __float_as_int(f);
  return (i >= 0) ? i : (i ^ 0x7FFFFFFF);
}
__device__ __forceinline__ float f32_unorder(int i) {
  return __int_as_float((i >= 0) ? i : (i ^ 0x7FFFFFFF));
}

// e = leaky_relu(es[src]+ed[dst]); segment max over dst (ordered-int atomicMax)
__global__ void edge_logits(const int* __restrict__ srcI, const int* __restrict__ dstI,
                            const float* __restrict__ es, const float* __restrict__ ed,
                            float* __restrict__ ew, int* __restrict__ m)
{
  int i = blockIdx.x * blockDim.x + threadIdx.x;
  if (i >= EPE) return;
  int s, d;
  if (i < EE) { s = srcI[i]; d = dstI[i]; } else { s = d = i - EE; }
  float e = es[s] + ed[d];
  e = (e > 0.0f) ? e : 0.2f * e;
  ew[i] = e;
  atomicMax(&m[d], f32_order(e));
}

// w = exp(e - m[dst]) (in place); denom[dst] += w
__global__ void edge_softmax_num(const int* __restrict__ dstI,
                                 const int* __restrict__ m,
                                 float* __restrict__ ew, float* __restrict__ denom)
{
  int i = blockIdx.x * blockDim.x + threadIdx.x;
  if (i >= EPE) return;
  int d = (i < EE) ? dstI[i] : (i - EE);
  float w = __expf(ew[i] - f32_unorder(m[d]));
  ew[i] = w;
  atomicAdd(&denom[d], w);
}

// out[dst] += (w/denom[dst]) * h[src] ; vectorized over OS (=round4(Cout)) cols.
// Pad columns of H are exactly 0, so they add 0 to the pad columns of out.
__global__ void edge_aggregate(const int* __restrict__ srcI, const int* __restrict__ dstI,
                               const float* __restrict__ ew, const float* __restrict__ denom,
                               const float* __restrict__ H, float* __restrict__ out,
                               int HS, int OS)
{
  int i = blockIdx.x * blockDim.x + threadIdx.x;
  if (i >= EPE) return;
  int s, d;
  if (i < EE) { s = srcI[i]; d = dstI[i]; } else { s = d = i - EE; }
  float alpha = ew[i] / denom[d];
  const float4* hs = (const float4*)(H + (long long)s * HS);  // 64B-aligned rows
  float* od = out + (long long)d * OS;
  for (int c4 = 0; c4 < OS / 4; ++c4) {
    float4 v = hs[c4];                                        // global_load_b128
    atomicAdd(&od[c4 * 4 + 0], alpha * v.x);
    atomicAdd(&od[c4 * 4 + 1], alpha * v.y);
    atomicAdd(&od[c4 * 4 + 2], alpha * v.z);
    atomicAdd(&od[c4 * 4 + 3], alpha * v.w);
  }
}

__global__ void relu_k(float* __restrict__ x, long long n)
{
  long long i = (long long)blockIdx.x * blockDim.x + threadIdx.x;
  if (i < n) x[i] = fmaxf(x[i], 0.0f);
}

// global_mean_pool: one wave per graph, lane = latent channel (LATENT == 32)
__global__ __launch_bounds__(32) void pool_mean(const float* __restrict__ x3,
                                                float* __restrict__ xg)
{
  int g = blockIdx.x, c = threadIdx.x;
  const float* base = x3 + (long long)g * NPG * 32;
  float s = 0.0f;
  for (int j = 0; j < NPG; ++j) s += base[j * 32 + c];
  xg[g * 32 + c] = s * (1.0f / (float)NPG);
}

// fc_expand -> x_exp stored at stride 4 with zero pad column
__global__ void fc_expand(const float* __restrict__ xg, const float* __restrict__ W,
                          const float* __restrict__ b, float* __restrict__ xe)
{
  int p = blockIdx.x * blockDim.x + threadIdx.x;    // over N*4
  if (p >= NN * 4) return;
  int n = p >> 2, c = p & 3;
  if (c == 3) { xe[p] = 0.0f; return; }
  int g = n / NPG, j = (n % NPG) * 3 + c;
  const float* xr = xg + g * 32;
  float s = b[j];
#pragma unroll
  for (int k = 0; k < 32; ++k) s += xr[k] * W[k * (NPG * 3) + j];
  xe[p] = s;
}

// tanh on cols 0,1; relu on col 2 (input at stride 4)
__global__ void final_act(const float* __restrict__ xr, float* __restrict__ out)
{
  int i = blockIdx.x * blockDim.x + threadIdx.x;    // over N*3
  if (i >= NN * 3) return;
  int n = i / 3, c = i % 3;
  float v = xr[n * 4 + c];
  out[i] = (c < 2) ? tanhf(v) : fmaxf(v, 0.0f);
}

// ============================================================================
static void launch_gemm(int Cin, int Cout, const float* X, const float* Wp,
                        float* H, hipStream_t s)
{
  dim3 g(NN / 64, round16(Cout) / 16);
  if      (Cin <= 4  && Cout == 32) gemm_wmma<4, 32><<<g, 32, 0, s>>>(X, Wp, H);
  else if (Cin <= 4  && Cout == 64) gemm_wmma<4, 64><<<g, 32, 0, s>>>(X, Wp, H);
  else if (Cin == 32 && Cout == 64) gemm_wmma<32, 64><<<g, 32, 0, s>>>(X, Wp, H);
  else if (Cin == 64 && Cout == 32) gemm_wmma<64, 32><<<g, 32, 0, s>>>(X, Wp, H);
  else /* Cin == 32 && Cout == 3 */ gemm_wmma<32, 16><<<g, 32, 0, s>>>(X, Wp, H);
}

struct GatParams { const float *W, *as, *ad, *b; int Cin, Cout; bool relu; };

static void run_gat(const GatParams& p, float* x, float* h, float* es, float* ed,
                    int* m, float* denom, float* ew, float* wpad,
                    const int* srcI, const int* dstI, hipStream_t stream)
{
  const int KP = round4(p.Cin), HS = round16(p.Cout), OS = round4(p.Cout);
  pack_weights<<<(KP * HS + 255) / 256, 256, 0, stream>>>(p.W, wpad, p.Cin, p.Cout, KP, HS);
  launch_gemm(p.Cin, p.Cout, x, wpad, h, stream);
  node_scores<<<(NN + 255) / 256, 256, 0, stream>>>(h, p.as, p.ad, es, ed, p.Cout, HS);
  long long nc = (long long)NN * OS;
  init_layer<<<(unsigned)((nc + 255) / 256), 256, 0, stream>>>(x, p.b, m, denom, p.Cout, OS);
  unsigned eg = (EPE + 255) / 256;
  edge_logits<<<eg, 256, 0, stream>>>(srcI, dstI, es, ed, ew, m);
  edge_softmax_num<<<eg, 256, 0, stream>>>(dstI, m, ew, denom);
  edge_aggregate<<<eg, 256, 0, stream>>>(srcI, dstI, ew, denom, h, x, HS, OS);
  if (p.relu) relu_k<<<(unsigned)((nc + 255) / 256), 256, 0, stream>>>(x, nc);
}

extern "C" void kernel_launch(void* const* d_in, const int* in_sizes, int n_in,
                              void* d_out, int out_size, void* d_ws, size_t ws_size,
                              hipStream_t stream)
{
  const float* x0   = (const float*)d_in[0];
  const int*   ei   = (const int*)d_in[1];
  const int*   srcI = ei;
  const int*   dstI = ei + EE;

  const float* e1W = (const float*)d_in[3];  const float* e1s = (const float*)d_in[4];
  const float* e1d = (const float*)d_in[5];  const float* e1b = (const float*)d_in[6];
  const float* e2W = (const float*)d_in[7];  const float* e2s = (const float*)d_in[8];
  const float* e2d = (const float*)d_in[9];  const float* e2b = (const float*)d_in[10];
  const float* e3W = (const float*)d_in[11]; const float* e3s = (const float*)d_in[12];
  const float* e3d = (const float*)d_in[13]; const float* e3b = (const float*)d_in[14];
  const float* fcW = (const float*)d_in[15]; const float* fcb = (const float*)d_in[16];
  const float* d1W = (const float*)d_in[17]; const float* d1s = (const float*)d_in[18];
  const float* d1d = (const float*)d_in[19]; const float* d1b = (const float*)d_in[20];
  const float* d2W = (const float*)d_in[21]; const float* d2s = (const float*)d_in[22];
  const float* d2d_ = (const float*)d_in[23]; const float* d2b = (const float*)d_in[24];
  const float* d3W = (const float*)d_in[25]; const float* d3s = (const float*)d_in[26];
  const float* d3d = (const float*)d_in[27]; const float* d3b = (const float*)d_in[28];

  // Workspace layout (floats)
  float* ws    = (float*)d_ws;
  float* bufA  = ws;                               // N*64 : x / aggregated out
  float* bufB  = bufA + (size_t)NN * 64;           // N*64 : h = x @ W (padded)
  float* es    = bufB + (size_t)NN * 64;           // N
  float* ed    = es + NN;                          // N
  int*   m     = (int*)(ed + NN);                  // N (ordered-int max)
  float* denom = (float*)(m + NN);                 // N
  float* ew    = denom + NN;                       // EPE (logit -> numerator)
  float* wpad  = ew + (size_t)EPE;                 // 64*64 padded weights

  float* outF = (float*)d_out;                     // [0, N*3) reconstruction
  float* xg   = outF + (size_t)NN * 3;             // x_global [B,32]

  // x -> bufA at stride 4, zero padded
  pack_x<<<(NN * 4 + 255) / 256, 256, 0, stream>>>(x0, bufA);

  // Encoder
  GatParams enc[3] = {
    { e1W, e1s, e1d, e1b,  3, 32, true },
    { e2W, e2s, e2d, e2b, 32, 64, true },
    { e3W, e3s, e3d, e3b, 64, 32, true },
  };
  for (int i = 0; i < 3; ++i)
    run_gat(enc[i], bufA, bufB, es, ed, m, denom, ew, wpad, srcI, dstI, stream);

  // global_mean_pool -> x_global (written straight into d_out region)
  pool_mean<<<BB, 32, 0, stream>>>(bufA, xg);

  // fc_expand -> x_exp in bufA ([N,4] zero padded)
  fc_expand<<<(NN * 4 + 255) / 256, 256, 0, stream>>>(xg, fcW, fcb, bufA);

  // Decoder
  GatParams dec[3] = {
    { d1W, d1s, d1d, d1b,  3, 64, true  },
    { d2W, d2s, d2d_, d2b, 64, 32, true },
    { d3W, d3s, d3d, d3b, 32,  3, false },
  };
  for (int i = 0; i < 3; ++i)
    run_gat(dec[i], bufA, bufB, es, ed, m, denom, ew, wpad, srcI, dstI, stream);

  // custom activation: tanh(cols 0..1), relu(col 2)
  final_act<<<(NN * 3 + 255) / 256, 256, 0, stream>>>(bufA, outF);
}